// GATDecoder_14405320311214
// MI455X (gfx1250) — compile-verified
//
#include <hip/hip_runtime.h>
#include <hip/hip_bf16.h>

#define NNODES 20000
#define NEDGES 640000
#define LATENT 64
#define HIDDEN 128
#define NFEAT  64

typedef __attribute__((ext_vector_type(2))) float v2f;
typedef __attribute__((ext_vector_type(8))) float v8f;

// ---------------------------------------------------------------------------
// FC GEMV: x0[row] = relu(dot(fc_w[row,0:64], z) + fc_b[row]),  row < N*LATENT
// One wave per row; lane reads float2 (coalesced 256B per wave), shuffle-reduce.
// ---------------------------------------------------------------------------
__global__ __launch_bounds__(256) void fc_relu_kernel(
    const float* __restrict__ z, const float* __restrict__ fcw,
    const float* __restrict__ fcb, float* __restrict__ x0, int rows) {
  __shared__ float sz[LATENT];
  const int tid = threadIdx.x;
  if (tid < LATENT) sz[tid] = z[tid];
  __syncthreads();
  const int lane = tid & 31;
  const int row = blockIdx.x * 8 + (tid >> 5);
  if (row >= rows) return;
  const float2 w = ((const float2*)(fcw + (size_t)row * LATENT))[lane];
  float p = w.x * sz[2 * lane] + w.y * sz[2 * lane + 1];
#pragma unroll
  for (int off = 16; off; off >>= 1) p += __shfl_down(p, off, 32);
  if (lane == 0) {
    float v = p + fcb[row];
    x0[row] = v > 0.f ? v : 0.f;
  }
}

// ---------------------------------------------------------------------------
// Node transform H = X @ W^T via V_WMMA_F32_16X16X4_F32 (full fp32 precision).
// Block: (FOUT/16) waves; each wave owns one 16-col strip; ROWT row-tiles/block.
// LDS padded so (FIN+PAD)%64==4 -> conflict-free 16-lane strided reads.
// ---------------------------------------------------------------------------
template <int FIN, int FOUT, int ROWT>
__global__ __launch_bounds__((FOUT / 16) * 32) void gemm_wmma_kernel(
    const float* __restrict__ X, const float* __restrict__ W,
    float* __restrict__ H) {
  constexpr int NW = FOUT / 16;
  constexpr int PAD = 4;  // (64+4)%64==4, (128+4)%64==4
  constexpr int LDW = FIN + PAD;
  __shared__ float sW[FOUT * LDW];
  __shared__ float sX[16 * ROWT * LDW];
  const int tid = threadIdx.x;
  const int wave = tid >> 5;
  const int lane = tid & 31;
  const int half = lane >> 4;
  const int l16 = lane & 15;
  const int rowBase = blockIdx.x * (16 * ROWT);

  for (int i = tid; i < FOUT * FIN; i += NW * 32)
    sW[(i / FIN) * LDW + (i % FIN)] = W[i];
  for (int i = tid; i < 16 * ROWT * FIN; i += NW * 32)
    sX[(i / FIN) * LDW + (i % FIN)] = X[(size_t)(rowBase + i / FIN) * FIN + (i % FIN)];
  __syncthreads();

  const int colBase = wave * 16;
  const float* wrow = &sW[(colBase + l16) * LDW];  // B: column n = colBase+l16
  for (int rt = 0; rt < ROWT; ++rt) {
    const float* xrow = &sX[(rt * 16 + l16) * LDW];  // A: row m = l16
    v8f c = {0.f, 0.f, 0.f, 0.f, 0.f, 0.f, 0.f, 0.f};
#pragma unroll
    for (int kb = 0; kb < FIN; kb += 4) {
      const int k = kb + 2 * half;  // VGPR0: K={0|2}, VGPR1: K={1|3}
      v2f a, b;
      a.x = xrow[k]; a.y = xrow[k + 1];
      b.x = wrow[k]; b.y = wrow[k + 1];  // W^T[k][n] = W[n][k]
      c = __builtin_amdgcn_wmma_f32_16x16x4_f32(false, a, false, b, (short)0, c,
                                                false, false);
    }
    const int node0 = rowBase + rt * 16 + half * 8;  // C: VGPR r -> M = 8*half+r
#pragma unroll
    for (int r = 0; r < 8; ++r)
      H[(size_t)(node0 + r) * FOUT + colBase + l16] = c[r];
  }
}

// ---------------------------------------------------------------------------
// Per-node attention scalars: ssrc[n] = h[n].a_s ; sdst[n] = h[n].a_d
// ---------------------------------------------------------------------------
template <int F>
__global__ __launch_bounds__(256) void node_dots_kernel(
    const float* __restrict__ H, const float* __restrict__ a_s,
    const float* __restrict__ a_d, float* __restrict__ ssrc,
    float* __restrict__ sdst, int n) {
  const int wid = (blockIdx.x * blockDim.x + threadIdx.x) >> 5;
  const int lane = threadIdx.x & 31;
  if (wid >= n) return;
  float ps = 0.f, pd = 0.f;
#pragma unroll
  for (int c = lane; c < F; c += 32) {
    const float h = H[(size_t)wid * F + c];
    ps += h * a_s[c];
    pd += h * a_d[c];
  }
#pragma unroll
  for (int off = 16; off; off >>= 1) {
    ps += __shfl_down(ps, off, 32);
    pd += __shfl_down(pd, off, 32);
  }
  if (lane == 0) { ssrc[wid] = ps; sdst[wid] = pd; }
}

// ---------------------------------------------------------------------------
// Init: acc = 0, m = -inf, s = 0
// ---------------------------------------------------------------------------
template <int F>
__global__ __launch_bounds__(256) void init_kernel(float* __restrict__ acc,
                                                   float* __restrict__ m,
                                                   float* __restrict__ s, int n) {
  const int i = blockIdx.x * blockDim.x + threadIdx.x;
  if (i < n) { m[i] = __int_as_float((int)0xff800000); s[i] = 0.f; }
  if (i < n * F) acc[i] = 0.f;
}

// float atomic max via monotone integer encoding (init must be -inf)
__device__ __forceinline__ void atomicMaxF32(float* addr, float val) {
  const int iv = __float_as_int(val);
  if (iv >= 0) atomicMax((int*)addr, iv);
  else atomicMin((unsigned int*)addr, (unsigned int)iv);
}

// ---------------------------------------------------------------------------
// Edge pass 1: e = leaky_relu(ssrc[src]+sdst[dst]); segment-max into m[dst]
// ---------------------------------------------------------------------------
__global__ __launch_bounds__(256) void edge_logits_max_kernel(
    const int* __restrict__ src, const int* __restrict__ dst,
    const float* __restrict__ ssrc, const float* __restrict__ sdst,
    float* __restrict__ e, float* __restrict__ m, int nE) {
  const int i = blockIdx.x * blockDim.x + threadIdx.x;
  if (i >= nE) return;
  float v = ssrc[src[i]] + sdst[dst[i]];
  v = v > 0.f ? v : 0.2f * v;
  e[i] = v;
  atomicMaxF32(&m[dst[i]], v);
}

// ---------------------------------------------------------------------------
// Edge pass 2: ex = exp(e - m[dst]); segment-sum into s[dst]
// ---------------------------------------------------------------------------
__global__ __launch_bounds__(256) void edge_exp_sum_kernel(
    const int* __restrict__ dst, const float* __restrict__ e,
    const float* __restrict__ m, float* __restrict__ ex, float* __restrict__ s,
    int nE) {
  const int i = blockIdx.x * blockDim.x + threadIdx.x;
  if (i >= nE) return;
  const int d = dst[i];
  const float v = expf(e[i] - m[d]);
  ex[i] = v;
  unsafeAtomicAdd(&s[d], v);  // global_atomic_add_f32, no return
}

// ---------------------------------------------------------------------------
// Edge pass 3: acc[dst] += h[src] * alpha  (wave per edge, lanes over Fout)
// ---------------------------------------------------------------------------
template <int F>
__global__ __launch_bounds__(256) void edge_scatter_kernel(
    const int* __restrict__ src, const int* __restrict__ dst,
    const float* __restrict__ ex, const float* __restrict__ ssum,
    const float* __restrict__ H, float* __restrict__ acc, int nE) {
  const int wid = (blockIdx.x * blockDim.x + threadIdx.x) >> 5;
  const int lane = threadIdx.x & 31;
  if (wid >= nE) return;
  const int sN = src[wid], dN = dst[wid];
  const float alpha = ex[wid] / (ssum[dN] + 1e-16f);
#pragma unroll
  for (int c = lane; c < F; c += 32)
    unsafeAtomicAdd(&acc[(size_t)dN * F + c], H[(size_t)sN * F + c] * alpha);
}

// ---------------------------------------------------------------------------
// out = (acc + b) with optional ReLU
// ---------------------------------------------------------------------------
template <int F, bool RELU>
__global__ __launch_bounds__(256) void bias_act_kernel(
    const float* __restrict__ acc, const float* __restrict__ b,
    float* __restrict__ out, int n) {
  const int i = blockIdx.x * blockDim.x + threadIdx.x;
  if (i >= n * F) return;
  float v = acc[i] + b[i % F];
  out[i] = RELU ? (v > 0.f ? v : 0.f) : v;
}

// ---------------------------------------------------------------------------
template <int FIN, int FOUT, bool RELU>
static void run_gat_layer(const float* x, const float* W, const float* a_s,
                          const float* a_d, const float* b, float* h, float* acc,
                          float* ssrc, float* sdst, float* m, float* s, float* e,
                          float* ex, const int* src, const int* dst, float* out,
                          hipStream_t stream) {
  constexpr int ROWT = 2;
  gemm_wmma_kernel<FIN, FOUT, ROWT>
      <<<NNODES / (16 * ROWT), (FOUT / 16) * 32, 0, stream>>>(x, W, h);
  node_dots_kernel<FOUT><<<(NNODES * 32 + 255) / 256, 256, 0, stream>>>(
      h, a_s, a_d, ssrc, sdst, NNODES);
  init_kernel<FOUT><<<(NNODES * FOUT + 255) / 256, 256, 0, stream>>>(acc, m, s,
                                                                     NNODES);
  edge_logits_max_kernel<<<(NEDGES + 255) / 256, 256, 0, stream>>>(
      src, dst, ssrc, sdst, e, m, NEDGES);
  edge_exp_sum_kernel<<<(NEDGES + 255) / 256, 256, 0, stream>>>(dst, e, m, ex, s,
                                                                NEDGES);
  edge_scatter_kernel<FOUT><<<(NEDGES * 32 + 255) / 256, 256, 0, stream>>>(
      src, dst, ex, s, h, acc, NEDGES);
  bias_act_kernel<FOUT, RELU><<<(NNODES * FOUT + 255) / 256, 256, 0, stream>>>(
      acc, b, out, NNODES);
}

extern "C" void kernel_launch(void* const* d_in, const int* in_sizes, int n_in,
                              void* d_out, int out_size, void* d_ws,
                              size_t ws_size, hipStream_t stream) {
  const float* z   = (const float*)d_in[0];
  const int*   ei  = (const int*)d_in[1];
  const float* fcw = (const float*)d_in[2];
  const float* fcb = (const float*)d_in[3];
  const float* W0  = (const float*)d_in[4];
  const float* as0 = (const float*)d_in[5];
  const float* ad0 = (const float*)d_in[6];
  const float* b0  = (const float*)d_in[7];
  const float* W1  = (const float*)d_in[8];
  const float* as1 = (const float*)d_in[9];
  const float* ad1 = (const float*)d_in[10];
  const float* b1  = (const float*)d_in[11];
  const float* W2  = (const float*)d_in[12];
  const float* as2 = (const float*)d_in[13];
  const float* ad2 = (const float*)d_in[14];
  const float* b2  = (const float*)d_in[15];

  const int* src = ei;            // edge_index[0]
  const int* dst = ei + NEDGES;   // edge_index[1]

  float* ws  = (float*)d_ws;
  float* xA  = ws;                                  // [N,128] (layer input; fc packs [N,64])
  float* hB  = xA + (size_t)NNODES * 128;           // [N,128]
  float* acc = hB + (size_t)NNODES * 128;           // [N,128]
  float* ssrc = acc + (size_t)NNODES * 128;         // [N]
  float* sdst = ssrc + NNODES;                      // [N]
  float* mbuf = sdst + NNODES;                      // [N]
  float* sbuf = mbuf + NNODES;                      // [N]
  float* ebuf = sbuf + NNODES;                      // [E]
  float* exbuf = ebuf + NEDGES;                     // [E]

  // x = relu(z @ fc_w.T + fc_b)  -> xA packed as [N, 64]
  fc_relu_kernel<<<(NNODES * LATENT) / 8, 256, 0, stream>>>(z, fcw, fcb, xA,
                                                            NNODES * LATENT);

  // layer 0: [N,64] -> [N,128], relu
  run_gat_layer<LATENT, HIDDEN, true>(xA, W0, as0, ad0, b0, hB, acc, ssrc, sdst,
                                      mbuf, sbuf, ebuf, exbuf, src, dst, xA,
                                      stream);
  // layer 1: [N,128] -> [N,128], relu
  run_gat_layer<HIDDEN, HIDDEN, true>(xA, W1, as1, ad1, b1, hB, acc, ssrc, sdst,
                                      mbuf, sbuf, ebuf, exbuf, src, dst, xA,
                                      stream);
  // layer 2: [N,128] -> [N,64], no relu, straight into d_out
  run_gat_layer<HIDDEN, NFEAT, false>(xA, W2, as2, ad2, b2, hB, acc, ssrc, sdst,
                                      mbuf, sbuf, ebuf, exbuf, src, dst,
                                      (float*)d_out, stream);
}